// LIF_38543036514405
// MI455X (gfx1250) — compile-verified
//
#include <hip/hip_runtime.h>
#include <stdint.h>

// Problem shape (fixed by the reference).
#define BB      32
#define TT_ALL  2048
#define NN      1024

// Tiling / pipeline.
#define LANES          32                    // one wave per block
#define TT             32                    // time steps per LDS tile
#define NBUF           7                     // ring depth for BOTH in and out rings
#define TILE_DW        (TT * LANES)          // 1024 dwords = 4 KB per tile
#define TILE_BYTES     (TILE_DW * 4)
#define ASYNC_PER_TILE 8                     // 8 x b128 (512 B/wave-instr) = 4 KB
#define ROWS_PER_ASYNC 4                     // 512 B / (32 lanes * 4 B) rows

// Wait until at most `pend` whole load-tiles can remain outstanding.
// ASYNCcnt counts loads+stores; loads <= total, and loads retire in order,
// so "total <= 8*pend" guarantees the oldest pending load tile has landed.
// It also bounds stores_outstanding <= 48 in steady state, which protects
// the 7-deep output ring (store batches retire in order among stores).
__device__ __forceinline__ void wait_async_tiles(int pend) {
    switch (pend) {
        case 0:  asm volatile("s_wait_asynccnt 0"  ::: "memory"); break;
        case 1:  asm volatile("s_wait_asynccnt 8"  ::: "memory"); break;
        case 2:  asm volatile("s_wait_asynccnt 16" ::: "memory"); break;
        case 3:  asm volatile("s_wait_asynccnt 24" ::: "memory"); break;
        case 4:  asm volatile("s_wait_asynccnt 32" ::: "memory"); break;
        case 5:  asm volatile("s_wait_asynccnt 40" ::: "memory"); break;
        default: asm volatile("s_wait_asynccnt 48" ::: "memory"); break;
    }
}

__global__ __launch_bounds__(LANES) void lif_scan_async_kernel(
    const float* __restrict__ in,      // [B, T, N]
    const float* __restrict__ thresh,  // [N]
    float* __restrict__ out)           // [B, T, N]
{
    constexpr int T = TT_ALL;
    constexpr int N = NN;
    constexpr int numTiles = T / TT;               // 64

    // [0 .. NBUF) tiles: input ring; [NBUF .. 2*NBUF): output ring. 56 KB total.
    __shared__ float smem[2 * NBUF * TILE_DW];

    const int lane    = threadIdx.x;               // 0..31
    const int nchunks = N / LANES;                 // 32
    const int b       = blockIdx.x / nchunks;
    const int nbase   = (blockIdx.x % nchunks) * LANES;

    // Lane-constant address pieces.
    // Each async b128: 32 lanes x 16 B = 512 B = 4 rows of this block's 128 B n-slice.
    // lane L -> row (L>>3) of the 4-row group, bytes (L&7)*16 within the row.
    const uint32_t lds_lane  = (uint32_t)(((lane >> 3) * LANES * 4) + (lane & 7) * 16);
    const uint32_t voff      = (uint32_t)(((lane >> 3) * N * 4) + nbase * 4 + (lane & 7) * 16);
    const uint32_t smem_base = (uint32_t)(uintptr_t)(&smem[0]); // low 32b of generic LDS ptr = LDS offset
    const uint32_t obuf_base = smem_base + (uint32_t)(NBUF * TILE_BYTES);
    const uint64_t gbase     = (uint64_t)(uintptr_t)in  + (uint64_t)b * T * N * 4u;
    const uint64_t obase     = (uint64_t)(uintptr_t)out + (uint64_t)b * T * N * 4u;

    // Issue one tile's 8 async b128 loads (global -> LDS input ring slot).
    auto issue_load_tile = [&](int tile, int slot) {
        const uint32_t lbase = smem_base + (uint32_t)slot * TILE_BYTES + lds_lane;
        const uint64_t sa    = gbase + (uint64_t)tile * ((uint64_t)TT * N * 4u);
        #pragma unroll
        for (int i = 0; i < ASYNC_PER_TILE; ++i) {
            asm volatile("global_load_async_to_lds_b128 %0, %1, %2"
                         :
                         : "v"(lbase + (uint32_t)(i * ROWS_PER_ASYNC * LANES * 4)),
                           "v"(voff),
                           "s"(sa + (uint64_t)(i * ROWS_PER_ASYNC) * N * 4u)
                         : "memory");
        }
    };

    // Issue one tile's 8 async b128 stores (LDS output ring slot -> global).
    auto issue_store_tile = [&](int tile, int slot) {
        const uint32_t lbase = obuf_base + (uint32_t)slot * TILE_BYTES + lds_lane;
        const uint64_t sa    = obase + (uint64_t)tile * ((uint64_t)TT * N * 4u);
        #pragma unroll
        for (int i = 0; i < ASYNC_PER_TILE; ++i) {
            asm volatile("global_store_async_from_lds_b128 %0, %1, %2"
                         :
                         : "v"(voff),
                           "v"(lbase + (uint32_t)(i * ROWS_PER_ASYNC * LANES * 4)),
                           "s"(sa + (uint64_t)(i * ROWS_PER_ASYNC) * N * 4u)
                         : "memory");
        }
    };

    // Prologue: fill the input ring.
    {
        const int pro = (numTiles < NBUF) ? numTiles : NBUF;
        int s = 0;
        for (int t = 0; t < pro; ++t) {
            issue_load_tile(t, s);
            if (++s == NBUF) s = 0;
        }
    }

    const float th = thresh[nbase + lane];
    float acc = 0.0f;

    int slot = 0;  // in-slot and out-slot cycle in lockstep
    for (int k = 0; k < numTiles; ++k) {
        // Ensure tile k's async loads have landed in LDS.
        const int rem_after = numTiles - 1 - k;
        wait_async_tiles(rem_after < (NBUF - 1) ? rem_after : (NBUF - 1));

        const float* ibuf = &smem[slot * TILE_DW];
        float*       obuf = &smem[(NBUF + slot) * TILE_DW];
        #pragma unroll
        for (int r = 0; r < TT; ++r) {
            const float x  = ibuf[r * LANES + lane];  // ds_load, conflict-free
            acc += x;
            const bool  sp = acc > th;
            const float o  = sp ? acc : 0.0f;
            acc            = sp ? 0.0f : acc;         // reset fired neurons
            obuf[r * LANES + lane] = o;               // ds_store, conflict-free
        }

        // Retire our ds_stores (producer for async store) and ds_loads
        // (WAR for the input-slot refill) before touching this slot again.
        asm volatile("s_wait_dscnt 0" ::: "memory");

        // Drain tile k's outputs LDS -> global (512 B per counter slot).
        issue_store_tile(k, slot);

        // Refill the input slot we just drained.
        if (k + NBUF < numTiles) {
            issue_load_tile(k + NBUF, slot);
        }
        if (++slot == NBUF) slot = 0;
    }
    // S_ENDPGM performs an implicit wait-idle: all async stores drain before exit.
}

extern "C" void kernel_launch(void* const* d_in, const int* in_sizes, int n_in,
                              void* d_out, int out_size, void* d_ws, size_t ws_size,
                              hipStream_t stream) {
    (void)in_sizes; (void)n_in; (void)d_ws; (void)ws_size; (void)out_size;
    const float* in     = (const float*)d_in[0];   // [32, 2048, 1024] f32
    const float* thresh = (const float*)d_in[1];   // [1024] f32
    float* out          = (float*)d_out;           // [32, 2048, 1024] f32

    const int nblocks = BB * (NN / LANES);         // 32 * 32 = 1024 single-wave blocks
    lif_scan_async_kernel<<<dim3(nblocks), dim3(LANES), 0, stream>>>(in, thresh, out);
}